// GraphEncoder_19232863552046
// MI455X (gfx1250) — compile-verified
//
#include <hip/hip_runtime.h>

typedef float v2f __attribute__((ext_vector_type(2)));
typedef float v8f __attribute__((ext_vector_type(8)));

#define IN_CH 512
#define HID   128
#define OUTC  64

// ---------------------------------------------------------------- degree prep
__global__ void k_set_ones(float* deg, int n) {
  int i = blockIdx.x * blockDim.x + threadIdx.x;
  if (i < n) deg[i] = 1.0f;                       // self-loop weight
}

__global__ void k_deg_accum(const int* __restrict__ ei, const float* __restrict__ ew,
                            float* deg, int E) {
  int e = blockIdx.x * blockDim.x + threadIdx.x;
  if (e < E) atomicAdd(&deg[ei[E + e]], ew[e]);   // dst-side degree
}

__global__ void k_rsqrt(const float* __restrict__ deg, float* dinv, int n) {
  int i = blockIdx.x * blockDim.x + threadIdx.x;
  if (i < n) {
    float d = deg[i];
    dinv[i] = (d > 0.0f) ? rsqrtf(d) : 0.0f;
  }
}

// ---------------------------------------------------------------- weight pack
// Repack row-major W[K,N] into WMMA-fragment order so each lane's B fragment
// is one contiguous float2:
//   Wp[((k4*2 + hi)*N + n)*2 + j] = W[(4*k4 + 2*hi + j)*N + n]
__global__ void k_pack_w(const float* __restrict__ W, float* __restrict__ Wp,
                         int K, int N) {
  int idx = blockIdx.x * blockDim.x + threadIdx.x;
  if (idx >= K * N) return;
  int k = idx / N;
  int n = idx - k * N;
  int k4 = k >> 2;
  int hi = (k >> 1) & 1;
  int j  = k & 1;
  Wp[(((size_t)((k4 << 1) | hi)) * N + n) * 2 + j] = W[idx];
}

// ---------------------------------------------------------------- WMMA GEMM
// C[M, NT*16] = A[M,K] @ B[K, NT*16] with B pre-packed by k_pack_w.
// One wave computes a 16 x (NT*16) tile with NT accumulators of
// V_WMMA_F32_16X16X4_F32. Requires M%16==0, K%8==0.
// k-loop is unrolled x2 with all fragments staged in distinct registers so
// the load batch of half-step 1 overlaps the WMMA group of half-step 0.
template <int NT>
__global__ void k_gemm_wmma(const float* __restrict__ A,
                            const float* __restrict__ Bp,
                            float* __restrict__ C,
                            int M, int K) {
  const int N = NT * 16;
  int wave = (int)((blockIdx.x * blockDim.x + threadIdx.x) >> 5);
  int lane = threadIdx.x & 31;
  if (wave >= (M >> 4)) return;

  int m0 = wave << 4;
  int lo = lane & 15;        // M index (A) / N index (B,C)
  int hi = lane >> 4;        // K pair select: {0,1} vs {2,3}

  const float* aPtr = A  + (size_t)(m0 + lo) * K + (hi << 1);
  const float* bPtr = Bp + ((size_t)hi * N + lo) * 2;

  v8f z = {0.f,0.f,0.f,0.f,0.f,0.f,0.f,0.f};
  v8f acc[NT];
#pragma unroll
  for (int t = 0; t < NT; ++t) acc[t] = z;

  for (int k = 0; k < K; k += 8) {
    // ---- stage all fragments for both half-steps (distinct registers) ----
    v2f a0 = *(const v2f*)(aPtr + k);
    v2f a1 = *(const v2f*)(aPtr + k + 4);
    const float* bp0 = bPtr + (size_t)k * N;
    const float* bp1 = bp0 + (size_t)4 * N;
    v2f b0[NT], b1[NT];
#pragma unroll
    for (int t = 0; t < NT; ++t) b0[t] = *(const v2f*)(bp0 + t * 32);
#pragma unroll
    for (int t = 0; t < NT; ++t) b1[t] = *(const v2f*)(bp1 + t * 32);
    // ---- compute: group 0 overlaps with group-1 loads still in flight ----
#pragma unroll
    for (int t = 0; t < NT; ++t)
      acc[t] = __builtin_amdgcn_wmma_f32_16x16x4_f32(
          false, a0, false, b0[t], (short)0, acc[t], false, false);
#pragma unroll
    for (int t = 0; t < NT; ++t)
      acc[t] = __builtin_amdgcn_wmma_f32_16x16x4_f32(
          false, a1, false, b1[t], (short)0, acc[t], false, false);
  }

  // C layout: VGPR r -> row m0 + 8*hi + r, col t*16 + lo
  int cRow = m0 + (hi << 3);
#pragma unroll
  for (int r = 0; r < 8; ++r) {
    float* cp = C + (size_t)(cRow + r) * N + lo;
#pragma unroll
    for (int t = 0; t < NT; ++t) cp[t * 16] = acc[t][r];
  }
}

// ------------------------------------------------- layer-1 init / scatter / relu
__global__ void k_init_agg1(const float* __restrict__ h1, const float* __restrict__ dinv,
                            const float* __restrict__ b1, float* agg, int n) {
  int idx = blockIdx.x * blockDim.x + threadIdx.x;
  if (idx < n) {
    int row = idx >> 7, c = idx & 127;
    float di = dinv[row];
    agg[idx] = h1[idx] * di * di + b1[c];         // self-loop + bias
  }
}

__global__ void k_scatter1(const int* __restrict__ ei, const float* __restrict__ ew,
                           const float* __restrict__ dinv, const float* __restrict__ h,
                           float* agg, int E) {
  long long idx = (long long)blockIdx.x * blockDim.x + threadIdx.x;
  if (idx >= (long long)E * 32) return;
  int e = (int)(idx >> 5);
  int g = ((int)idx & 31) << 2;                   // 4-channel group, C=128
  int s = ei[e], d = ei[E + e];
  float nm = dinv[s] * ew[e] * dinv[d];
  float4 v = *(const float4*)(h + (size_t)s * HID + g);
  float* o = agg + (size_t)d * HID + g;
  atomicAdd(o + 0, v.x * nm);
  atomicAdd(o + 1, v.y * nm);
  atomicAdd(o + 2, v.z * nm);
  atomicAdd(o + 3, v.w * nm);
}

__global__ void k_relu(float* a, int n) {
  int i = blockIdx.x * blockDim.x + threadIdx.x;
  if (i < n) a[i] = fmaxf(a[i], 0.0f);
}

// ------------------------------------------------- layer-2 init / scatter
__global__ void k_init_out(const float* __restrict__ t2, const float* __restrict__ t3,
                           const float* __restrict__ dinv,
                           const float* __restrict__ b2, const float* __restrict__ b3,
                           float* om, float* ol, int n) {
  int idx = blockIdx.x * blockDim.x + threadIdx.x;
  if (idx < n) {
    int row = idx >> 6, c = idx & 63;
    float di = dinv[row];
    float d2 = di * di;
    om[idx] = t2[idx] * d2 + b2[c];
    ol[idx] = t3[idx] * d2 + b3[c];
  }
}

__global__ void k_scatter2(const int* __restrict__ ei, const float* __restrict__ ew,
                           const float* __restrict__ dinv,
                           const float* __restrict__ t2, const float* __restrict__ t3,
                           float* om, float* ol, int E) {
  long long idx = (long long)blockIdx.x * blockDim.x + threadIdx.x;
  if (idx >= (long long)E * 32) return;
  int e = (int)(idx >> 5);
  int l = (int)idx & 31;                          // 0-15 -> mean, 16-31 -> logstd
  int s = ei[e], d = ei[E + e];
  float nm = dinv[s] * ew[e] * dinv[d];
  const float* tin = (l < 16) ? t2 : t3;
  float*       op  = (l < 16) ? om : ol;
  int g = (l & 15) << 2;                          // 4-channel group, C=64
  float4 v = *(const float4*)(tin + (size_t)s * OUTC + g);
  float* o = op + (size_t)d * OUTC + g;
  atomicAdd(o + 0, v.x * nm);
  atomicAdd(o + 1, v.y * nm);
  atomicAdd(o + 2, v.z * nm);
  atomicAdd(o + 3, v.w * nm);
}

// ---------------------------------------------------------------- launcher
extern "C" void kernel_launch(void* const* d_in, const int* in_sizes, int n_in,
                              void* d_out, int out_size, void* d_ws, size_t ws_size,
                              hipStream_t stream) {
  const float* x  = (const float*)d_in[0];
  const int*   ei = (const int*)  d_in[1];
  const float* ew = (const float*)d_in[2];
  const float* W1 = (const float*)d_in[3];
  const float* b1 = (const float*)d_in[4];
  const float* W2 = (const float*)d_in[5];
  const float* b2 = (const float*)d_in[6];
  const float* W3 = (const float*)d_in[7];
  const float* b3 = (const float*)d_in[8];

  const int N = in_sizes[0] / IN_CH;   // 50000
  const int E = in_sizes[2];           // 1600000

  // workspace layout (floats):
  //   deg[N] | dinv[N] | h1[N*128] | agg1[N*128] | W1p[512*128] | W2p[128*64] | W3p[128*64]
  // t2/t3 reuse the h1 region once layer-1 is folded into agg1.
  float* ws   = (float*)d_ws;
  float* deg  = ws;
  float* dinv = ws + N;
  float* h1   = ws + 2 * (size_t)N;
  float* agg1 = h1 + (size_t)N * HID;            // becomes h after relu
  float* W1p  = agg1 + (size_t)N * HID;
  float* W2p  = W1p + (size_t)IN_CH * HID;
  float* W3p  = W2p + (size_t)HID * OUTC;
  float* t2   = h1;                               // reuse (h1 dead after scatter1)
  float* t3   = h1 + (size_t)N * OUTC;
  float* om   = (float*)d_out;                    // mean   [N*64]
  float* ol   = om + (size_t)N * OUTC;            // logstd [N*64]

  const int T = 256;

  // 0) pack weights into WMMA fragment order
  k_pack_w<<<(IN_CH * HID + T - 1) / T, T, 0, stream>>>(W1, W1p, IN_CH, HID);
  k_pack_w<<<(HID * OUTC + T - 1) / T, T, 0, stream>>>(W2, W2p, HID, OUTC);
  k_pack_w<<<(HID * OUTC + T - 1) / T, T, 0, stream>>>(W3, W3p, HID, OUTC);

  // 1) degree (self-loop included) and dinv
  k_set_ones  <<<(N + T - 1) / T, T, 0, stream>>>(deg, N);
  k_deg_accum <<<(E + T - 1) / T, T, 0, stream>>>(ei, ew, deg, E);
  k_rsqrt     <<<(N + T - 1) / T, T, 0, stream>>>(deg, dinv, N);

  // 2) h1 = x @ W1  (WMMA fp32, 16x128 tile per wave)
  {
    int waves  = N >> 4;                          // 3125
    int blocks = (waves * 32 + T - 1) / T;
    k_gemm_wmma<8><<<blocks, T, 0, stream>>>(x, W1p, h1, N, IN_CH);
  }

  // 3) layer-1 aggregate: agg1 = h1*dinv^2 + b1, scatter edges, relu -> h
  k_init_agg1<<<((N * HID) + T - 1) / T, T, 0, stream>>>(h1, dinv, b1, agg1, N * HID);
  {
    long long tot = (long long)E * 32;
    int blocks = (int)((tot + T - 1) / T);
    k_scatter1<<<blocks, T, 0, stream>>>(ei, ew, dinv, h1, agg1, E);
  }
  k_relu<<<((N * HID) + T - 1) / T, T, 0, stream>>>(agg1, N * HID);
  const float* h = agg1;

  // 4) t2 = h @ W2, t3 = h @ W3  (WMMA fp32, 16x64 tile per wave)
  {
    int waves  = N >> 4;                          // 3125
    int blocks = (waves * 32 + T - 1) / T;
    k_gemm_wmma<4><<<blocks, T, 0, stream>>>(h, W2p, t2, N, HID);
    k_gemm_wmma<4><<<blocks, T, 0, stream>>>(h, W3p, t3, N, HID);
  }

  // 5) layer-2 aggregate directly into d_out
  k_init_out<<<((N * OUTC) + T - 1) / T, T, 0, stream>>>(t2, t3, dinv, b2, b3, om, ol, N * OUTC);
  {
    long long tot = (long long)E * 32;
    int blocks = (int)((tot + T - 1) / T);
    k_scatter2<<<blocks, T, 0, stream>>>(ei, ew, dinv, t2, t3, om, ol, E);
  }
}